// HorizonPredictor_3925600108905
// MI455X (gfx1250) — compile-verified
//
#include <hip/hip_runtime.h>
#include <math.h>

typedef __attribute__((ext_vector_type(2))) float v2f;
typedef __attribute__((ext_vector_type(8))) float v8f;

namespace {
constexpr int kBatch        = 32;
constexpr int kSeq          = 4096;
constexpr int kD            = 256;
constexpr int kH            = 8;
constexpr int kT            = 28;
constexpr int kRows         = kBatch * kSeq;       // 131072
constexpr int kWavesPerBlk  = 8;
constexpr int kRowsPerWave  = 32;                  // two 16-row WMMA tiles
constexpr int kRowsPerBlk   = kWavesPerBlk * kRowsPerWave;  // 256
constexpr int kThreads      = kWavesPerBlk * 32;   // 256
constexpr int kTPad         = 17;                  // transpose row stride (floats)
}

__device__ __forceinline__ float sigmoidf_(float x) {
  return 1.0f / (1.0f + __expf(-x));
}

__global__ __launch_bounds__(kThreads)
void horizon_kernel(const float* __restrict__ x,          // [rows, 256]
                    const float* __restrict__ last_step,  // [rows]
                    const float* __restrict__ W_ih,       // [24] (column 0)
                    const float* __restrict__ W_hh,       // [24,8]
                    const float* __restrict__ b_ih,       // [24]
                    const float* __restrict__ b_hh,       // [24]
                    const float* __restrict__ Wp,         // [8,256]
                    const float* __restrict__ bp,         // [8]
                    const float* __restrict__ Wo,         // [8]
                    const float* __restrict__ bo,         // [1]
                    const float* __restrict__ Wg1,        // [8,256]
                    const float* __restrict__ bg1,        // [8]
                    const float* __restrict__ Wg2,        // [28,8]
                    const float* __restrict__ bg2,        // [28]
                    const float* __restrict__ log_decay,  // [1]
                    float* __restrict__ out)              // [rows, 28]
{
  // Combined projection weight: rows 0..7 = Wp, rows 8..15 = Wg1 (each 256 wide)
  __shared__ float sWc[16 * kD];                 // 16 KB
  __shared__ float sWih[24], sBih[24], sBhh[24];
  __shared__ float sWhh[24 * 8];
  __shared__ float sWo[8], sBp[8], sBg1[8];
  __shared__ float sWg2[28 * 8], sBg2[28], sDecay[28];
  __shared__ float sBo[1];
  __shared__ float sT[kWavesPerBlk][kRowsPerWave * kTPad];  // per-wave transpose buf

  const int tid = threadIdx.x;

  // ---- stage weights in LDS (one pass per block) ----
  for (int i = tid; i < kH * kD; i += kThreads) {   // 2048 each, exact multiple
    sWc[i]           = Wp[i];
    sWc[kH * kD + i] = Wg1[i];
  }
  if (tid < 24) { sWih[tid] = W_ih[tid]; sBih[tid] = b_ih[tid]; sBhh[tid] = b_hh[tid]; }
  if (tid < 24 * 8) sWhh[tid] = W_hh[tid];
  if (tid < 8) { sWo[tid] = Wo[tid]; sBp[tid] = bp[tid]; sBg1[tid] = bg1[tid]; }
  if (tid == 0) sBo[0] = bo[0];
  if (tid < 28 * 8) sWg2[tid] = Wg2[tid];
  if (tid < 28) {
    sBg2[tid]   = bg2[tid];
    sDecay[tid] = __expf(-__expf(log_decay[0]) * (float)(tid + 1));
  }
  __syncthreads();

  // ---- WMMA projection: D(16x16) += A(16x4) * B(4x16), K-chain over 256 ----
  const int wave = tid >> 5;
  const int lane = tid & 31;
  const int mrow = lane & 15;          // A: row in tile / B: output column n
  const int koff = (lane >> 4) * 2;    // K sub-offset within the 4-wide step

  const long baseRow = (long)blockIdx.x * kRowsPerBlk + (long)wave * kRowsPerWave;
  const float* xr0 = x + (baseRow + mrow) * kD;         // tile 0: rows base..base+15
  const float* xr1 = x + (baseRow + 16 + mrow) * kD;    // tile 1: rows base+16..base+31
  const float* bsrc = &sWc[mrow * kD];                  // Wc row n

  v8f acc0 = {0.f,0.f,0.f,0.f,0.f,0.f,0.f,0.f};
  v8f acc1 = {0.f,0.f,0.f,0.f,0.f,0.f,0.f,0.f};
  for (int k = 0; k < kD; k += 4) {
    v2f a0 = *(const v2f*)(xr0 + k + koff);
    v2f a1 = *(const v2f*)(xr1 + k + koff);
    v2f bb = *(const v2f*)(bsrc + k + koff);
    acc0 = __builtin_amdgcn_wmma_f32_16x16x4_f32(false, a0, false, bb, (short)0,
                                                 acc0, false, false);
    acc1 = __builtin_amdgcn_wmma_f32_16x16x4_f32(false, a1, false, bb, (short)0,
                                                 acc1, false, false);
  }

  // ---- transpose 2x(16x16) tiles through LDS so each lane owns one row ----
  {
    float* tp = sT[wave];
    const int half = (lane >> 4) * 8;  // lanes 16..31 hold rows M = v+8
    #pragma unroll
    for (int v = 0; v < 8; ++v) {
      tp[(v + half) * kTPad + mrow]        = acc0[v];
      tp[(16 + v + half) * kTPad + mrow]   = acc1[v];
    }
  }
  __syncthreads();

  // ---- per-row scalar phase: lane owns row (baseRow + lane) ----
  const float* myrow = &sT[wave][lane * kTPad];
  float h[kH], g1[kH];
  #pragma unroll
  for (int j = 0; j < kH; ++j) {
    h[j]  = myrow[j] + sBp[j];
    float g = myrow[kH + j] + sBg1[j];
    g1[j] = g > 0.0f ? g : 0.0f;       // relu
  }

  const long row  = baseRow + lane;
  const float last = last_step[row];
  float cur = last;
  float* orow = out + row * kT;

  for (int t = 0; t < kT; ++t) {
    // r gates (need them before the candidate)
    float rr[kH];
    #pragma unroll
    for (int c = 0; c < kH; ++c) {
      float gh = sBhh[c];
      #pragma unroll
      for (int j = 0; j < kH; ++j) gh = fmaf(sWhh[c * kH + j], h[j], gh);
      float gi = fmaf(cur, sWih[c], sBih[c]);
      rr[c] = sigmoidf_(gi + gh);
    }
    // z, candidate, state update
    float hn[kH];
    #pragma unroll
    for (int c = 0; c < kH; ++c) {
      float ghz = sBhh[kH + c];
      float ghn = sBhh[2 * kH + c];
      #pragma unroll
      for (int j = 0; j < kH; ++j) {
        ghz = fmaf(sWhh[(kH + c) * kH + j], h[j], ghz);
        ghn = fmaf(sWhh[(2 * kH + c) * kH + j], h[j], ghn);
      }
      float z  = sigmoidf_(fmaf(cur, sWih[kH + c], sBih[kH + c]) + ghz);
      float nc = tanhf(fmaf(cur, sWih[2 * kH + c], sBih[2 * kH + c]) + rr[c] * ghn);
      hn[c] = (1.0f - z) * nc + z * h[c];
    }
    #pragma unroll
    for (int c = 0; c < kH; ++c) h[c] = hn[c];

    float pred = sBo[0];
    #pragma unroll
    for (int j = 0; j < kH; ++j) pred = fmaf(sWo[j], h[j], pred);
    cur = pred;

    float gp = sBg2[t];
    #pragma unroll
    for (int j = 0; j < kH; ++j) gp = fmaf(sWg2[t * kH + j], g1[j], gp);
    float gate = sigmoidf_(gp);

    orow[t] = gate * pred + (1.0f - gate) * last * sDecay[t];
  }
}

extern "C" void kernel_launch(void* const* d_in, const int* in_sizes, int n_in,
                              void* d_out, int out_size, void* d_ws, size_t ws_size,
                              hipStream_t stream) {
  (void)in_sizes; (void)n_in; (void)d_ws; (void)ws_size; (void)out_size;
  const float* x         = (const float*)d_in[0];
  const float* last_step = (const float*)d_in[1];
  const float* W_ih      = (const float*)d_in[2];
  const float* W_hh      = (const float*)d_in[3];
  const float* b_ih      = (const float*)d_in[4];
  const float* b_hh      = (const float*)d_in[5];
  const float* Wp        = (const float*)d_in[6];
  const float* bp        = (const float*)d_in[7];
  const float* Wo        = (const float*)d_in[8];
  const float* bo        = (const float*)d_in[9];
  const float* Wg1       = (const float*)d_in[10];
  const float* bg1       = (const float*)d_in[11];
  const float* Wg2       = (const float*)d_in[12];
  const float* bg2       = (const float*)d_in[13];
  const float* log_decay = (const float*)d_in[14];

  dim3 grid(kRows / kRowsPerBlk);   // 512
  dim3 block(kThreads);             // 256 = 8 wave32
  horizon_kernel<<<grid, block, 0, stream>>>(x, last_step, W_ih, W_hh, b_ih, b_hh,
                                             Wp, bp, Wo, bo, Wg1, bg1, Wg2, bg2,
                                             log_decay, (float*)d_out);
}